// S_GC_att_edge_5342939316459
// MI455X (gfx1250) — compile-verified
//
#include <hip/hip_runtime.h>
#include <hip/hip_bf16.h>

typedef __attribute__((ext_vector_type(2))) float v2f;
typedef __attribute__((ext_vector_type(8))) float v8f;

// Problem constants (fixed by the reference)
#define C_IN   256
#define T_DIM  300
#define V_DIM  25
#define O_DIM  256
#define K_TOT  7
#define S_KER  3

// LDS layout (floats)
#define XS_STRIDE 40                    // 2*40 mod 64 == 16 -> half-waves hit disjoint bank groups
#define XS_FLOATS (C_IN * XS_STRIDE)    // 10240
#define YS_STRIDE 257                   // odd stride -> conflict-free transposed D stores
#define YS_FLOATS (32 * YS_STRIDE)      // 8224
#define BS_FLOATS (K_TOT * V_DIM * V_DIM) // 4375
#define BIAS_FLOATS (K_TOT * O_DIM)       // 1792
#define SMEM_FLOATS (XS_FLOATS + YS_FLOATS + BS_FLOATS + BIAS_FLOATS)

__global__ __launch_bounds__(256)
void sgc_fused_wmma(const float* __restrict__ x,
                    const float* __restrict__ A,
                    const float* __restrict__ att,
                    const float* __restrict__ w,
                    const float* __restrict__ bias,
                    float* __restrict__ out) {
    extern __shared__ float smem[];
    float* xs  = smem;                          // [256][40]  x[n,:,t,:] (v padded w/ zeros)
    float* ys  = smem + XS_FLOATS;              // [32][257]  y_k transposed: ys[v][row]
    float* Bsh = ys + YS_FLOATS;                // [7][25][25] adjacency (static + per-n att)
    float* bsh = Bsh + BS_FLOATS;               // [1792] bias

    const int t    = blockIdx.x;
    const int n    = blockIdx.y;
    const int tid  = threadIdx.x;
    const int lane = tid & 31;
    const int wave = tid >> 5;
    const int hi   = lane >> 4;                 // 0 for lanes 0-15, 1 for lanes 16-31
    const int lo   = lane & 15;

    // ---- stage 0: fill LDS -------------------------------------------------
    const float* xbase = x + ((size_t)n * C_IN) * (T_DIM * V_DIM) + (size_t)t * V_DIM;
    for (int i = tid; i < C_IN * V_DIM; i += 256) {
        int c = i / V_DIM, v = i - c * V_DIM;
        xs[c * XS_STRIDE + v] = xbase[(size_t)c * (T_DIM * V_DIM) + v];
    }
    for (int i = tid; i < C_IN * (XS_STRIDE - V_DIM); i += 256) {     // zero pad v=25..39
        int c = i / (XS_STRIDE - V_DIM), v = V_DIM + i % (XS_STRIDE - V_DIM);
        xs[c * XS_STRIDE + v] = 0.0f;
    }
    for (int i = tid; i < S_KER * V_DIM * V_DIM; i += 256) Bsh[i] = A[i];
    for (int i = tid; i < (K_TOT - S_KER) * V_DIM * V_DIM; i += 256)
        Bsh[S_KER * V_DIM * V_DIM + i] =
            att[(size_t)n * (K_TOT - S_KER) * V_DIM * V_DIM + i];
    for (int i = tid; i < K_TOT * O_DIM; i += 256) bsh[i] = bias[i];
    __syncthreads();

    float acc[V_DIM];
    #pragma unroll
    for (int i = 0; i < V_DIM; ++i) acc[i] = 0.0f;

    // ---- per-k chunk: WMMA conv then VALU graph-mix ------------------------
    for (int k = 0; k < K_TOT; ++k) {
        const int rowb = k * O_DIM;             // base row in conv_w / bias

        // C/D 16x16 f32 layout: element j of lane l holds (M = j + 8*(l/16), N = l%16).
        // Seed accumulators with the bias (depends on M row only).
        v8f d[2][2];
        #pragma unroll
        for (int m = 0; m < 2; ++m) {
            const int mt = wave * 2 + m;
            #pragma unroll
            for (int nt = 0; nt < 2; ++nt) {
                #pragma unroll
                for (int j = 0; j < 8; ++j)
                    d[m][nt][j] = bsh[rowb + mt * 16 + j + 8 * hi];
            }
        }

        // A 16x4 f32 layout: lane l, vgpr j holds (M = l%16, K = 2*(l/16) + j)
        const float* wrow0 = w + (size_t)(rowb + wave * 32 + lo) * C_IN;

        #pragma unroll 4
        for (int k0 = 0; k0 < C_IN; k0 += 4) {
            // B 4x16 f32 layout: lane l, vgpr j holds (K = 2*(l/16) + j, N = l%16)
            v2f bfr[2];
            #pragma unroll
            for (int nt = 0; nt < 2; ++nt) {
                bfr[nt][0] = xs[(k0 + 2 * hi + 0) * XS_STRIDE + nt * 16 + lo];
                bfr[nt][1] = xs[(k0 + 2 * hi + 1) * XS_STRIDE + nt * 16 + lo];
            }
            #pragma unroll
            for (int m = 0; m < 2; ++m) {
                v2f afr = *(const v2f*)(wrow0 + (size_t)m * 16 * C_IN + k0 + 2 * hi);
                #pragma unroll
                for (int nt = 0; nt < 2; ++nt)
                    d[m][nt] = __builtin_amdgcn_wmma_f32_16x16x4_f32(
                        false, afr, false, bfr[nt], (short)0, d[m][nt], false, false);
            }
        }

        // Store D transposed into LDS: ys[v][row_local]
        #pragma unroll
        for (int m = 0; m < 2; ++m) {
            const int mt = wave * 2 + m;
            #pragma unroll
            for (int nt = 0; nt < 2; ++nt) {
                const int v = nt * 16 + lo;
                #pragma unroll
                for (int j = 0; j < 8; ++j)
                    ys[v * YS_STRIDE + mt * 16 + j + 8 * hi] = d[m][nt][j];
            }
        }
        __syncthreads();

        // Graph mix: thread tid owns output channel c=tid
        const float* Bk = Bsh + k * V_DIM * V_DIM;
        #pragma unroll 5
        for (int v = 0; v < V_DIM; ++v) {
            const float yv = ys[v * YS_STRIDE + tid];
            #pragma unroll
            for (int ww = 0; ww < V_DIM; ++ww)
                acc[ww] = fmaf(yv, Bk[v * V_DIM + ww], acc[ww]);
        }
        __syncthreads();                         // ys reused next k
    }

    float* op = out + (((size_t)n * O_DIM + tid) * T_DIM + t) * V_DIM;
    #pragma unroll
    for (int ww = 0; ww < V_DIM; ++ww) op[ww] = acc[ww];
}

extern "C" void kernel_launch(void* const* d_in, const int* in_sizes, int n_in,
                              void* d_out, int out_size, void* d_ws, size_t ws_size,
                              hipStream_t stream) {
    const float* x    = (const float*)d_in[0];   // [32,256,300,25]
    const float* A    = (const float*)d_in[1];   // [3,25,25]
    const float* att  = (const float*)d_in[2];   // [32,4,25,25]
    const float* w    = (const float*)d_in[3];   // [1792,256]
    const float* bias = (const float*)d_in[4];   // [1792]
    float* out        = (float*)d_out;           // [32,256,300,25]

    dim3 grid(T_DIM, 32);                        // one block per (t, n)
    size_t shmem = (size_t)SMEM_FLOATS * sizeof(float);  // ~96 KB (< 320 KB/WGP)
    sgc_fused_wmma<<<grid, dim3(256), shmem, stream>>>(x, A, att, w, bias, out);
}